// GroupedEmbeddingBag_71253507440830
// MI455X (gfx1250) — compile-verified
//
#include <hip/hip_runtime.h>
#include <hip/hip_bf16.h>
#include <stdint.h>

// Problem constants (from the reference):
#define T_TABLES 8
#define V_ROWS   100000
#define D_DIM    128        // 32 lanes x 4 floats
#define B_BAGS   4096
#define N_VALS   204800
#define CHUNK    32         // (index,weight) elements staged per async copy

typedef __attribute__((ext_vector_type(4))) float v4f;
typedef __attribute__((ext_vector_type(4))) int   v4i;

#if defined(__AMDGCN__) && __has_builtin(__builtin_amdgcn_global_load_async_to_lds_b32)
#define HAVE_ASYNC_LDS 1
#else
#define HAVE_ASYNC_LDS 0
#endif

// Builtin signature (from hipcc diagnostic): arg0 = global (AS1) int*,
// arg1 = LDS (AS3) int*, then imm offset, imm cpol.
typedef __attribute__((address_space(1))) int gint_t;   // global int
typedef __attribute__((address_space(3))) int lint_t;   // LDS int

// One wave32 per (table, bag). Lane l owns elements [4l,4l+4) of the 128-dim
// row -> each gathered row is one coalesced global_load_b128 per lane.
// Waves are table-major so co-resident blocks share one ~51MB table in the
// 192MB L2. Per wave, (index,weight) streams are double-buffered into LDS
// with GLOBAL_LOAD_ASYNC_TO_LDS_B32 (ASYNCcnt) so the hot loop's only global
// traffic is the B128 row gathers; indices come from low-latency ds_load.
__global__ __launch_bounds__(256) void grouped_embedding_bag_kernel(
    const float* __restrict__ tables,   // [T, V, D]
    const int*   __restrict__ values,   // [T, N]
    const int*   __restrict__ offsets,  // [T, B+1]
    const float* __restrict__ psw,      // [T, N]
    float*       __restrict__ out)      // [B, T*D]
{
    __shared__ alignas(16) int   s_idx[8][2][CHUNK];   // [wave][buf][elem]
    __shared__ alignas(16) float s_w  [8][2][CHUNK];

    const int lane = threadIdx.x & 31;
    const int wv   = threadIdx.x >> 5;
    const int wave = (blockIdx.x << 3) + wv;     // 8 waves / block
    const int t    = wave >> 12;                 // wave / B_BAGS (table-major)
    const int b    = wave & (B_BAGS - 1);

    const int obase = t * (B_BAGS + 1) + b;
    const int begin = offsets[obase];
    const int end   = offsets[obase + 1];
    const int len   = end - begin;

    const float* __restrict__ tab0 = tables + (size_t)t * ((size_t)V_ROWS * D_DIM);
    const float* __restrict__ tabL = tab0 + ((unsigned)lane << 2);
    const int*   __restrict__ vv   = values + (size_t)t * N_VALS;
    const float* __restrict__ ww   = psw    + (size_t)t * N_VALS;

    v4f acc = (v4f){0.f, 0.f, 0.f, 0.f};
    const int nch = (len + CHUNK - 1) >> 5;

    // Stage chunk 0 into buffer 0 (lane-parallel, EXEC-masked at the tail).
    if (nch > 0) {
        const int p = begin + lane;
        if (p < end) {
#if HAVE_ASYNC_LDS
            __builtin_amdgcn_global_load_async_to_lds_b32(
                (gint_t*)(vv + p), (lint_t*)&s_idx[wv][0][lane], 0, 0);
            __builtin_amdgcn_global_load_async_to_lds_b32(
                (gint_t*)(ww + p), (lint_t*)&s_w[wv][0][lane], 0, 0);
#else
            s_idx[wv][0][lane] = vv[p];
            s_w  [wv][0][lane] = ww[p];
#endif
        }
    }

    for (int c = 0; c < nch; ++c) {
        const int buf = c & 1;
        if (c + 1 < nch) {
            // Stage next chunk into the other buffer, then wait for current:
            // async copies complete in order, so <=2 outstanding means chunk
            // c's two copies have landed in LDS.
            const int p = begin + ((c + 1) << 5) + lane;
            if (p < end) {
#if HAVE_ASYNC_LDS
                __builtin_amdgcn_global_load_async_to_lds_b32(
                    (gint_t*)(vv + p), (lint_t*)&s_idx[wv][buf ^ 1][lane], 0, 0);
                __builtin_amdgcn_global_load_async_to_lds_b32(
                    (gint_t*)(ww + p), (lint_t*)&s_w[wv][buf ^ 1][lane], 0, 0);
#else
                s_idx[wv][buf ^ 1][lane] = vv[p];
                s_w  [wv][buf ^ 1][lane] = ww[p];
#endif
            }
#if HAVE_ASYNC_LDS
            asm volatile("s_wait_asynccnt 0x2" ::: "memory");
#endif
        } else {
#if HAVE_ASYNC_LDS
            asm volatile("s_wait_asynccnt 0x0" ::: "memory");
#endif
        }
#if !HAVE_ASYNC_LDS
        __builtin_amdgcn_wave_barrier();   // wave32 in-order DS: pin ordering
#endif

        const int count = min(CHUNK, len - (c << 5));

        // Lane-parallel prefetch: lane l pulls all 4x128B lines of row l of
        // this chunk into cache (4 global_prefetch_b8 cover 32 rows = 16KB).
        if (lane < count) {
            const float* rp = tab0 + (size_t)s_idx[wv][buf][lane] * D_DIM;
            __builtin_prefetch(rp,      0, 1);
            __builtin_prefetch(rp + 32, 0, 1);
            __builtin_prefetch(rp + 64, 0, 1);
            __builtin_prefetch(rp + 96, 0, 1);
        }

        int j = 0;
        for (; j + 4 <= count; j += 4) {
            // Broadcast ds_load_b128: 4 indices + 4 weights for the wave.
            const v4i id = *(const v4i*)&s_idx[wv][buf][j];
            const v4f wt = *(const v4f*)&s_w  [wv][buf][j];
            const v4f r0 = *(const v4f*)(tabL + (size_t)id.x * D_DIM);
            const v4f r1 = *(const v4f*)(tabL + (size_t)id.y * D_DIM);
            const v4f r2 = *(const v4f*)(tabL + (size_t)id.z * D_DIM);
            const v4f r3 = *(const v4f*)(tabL + (size_t)id.w * D_DIM);
            acc += r0 * wt.x;
            acc += r1 * wt.y;
            acc += r2 * wt.z;
            acc += r3 * wt.w;
        }
        for (; j < count; ++j) {    // chunk tail (0..3 rows)
            const v4f r = *(const v4f*)(tabL + (size_t)s_idx[wv][buf][j] * D_DIM);
            acc += r * s_w[wv][buf][j];
        }
    }

    // out[b, t*D + 4*lane ..] -- write-once: non-temporal B128 store keeps the
    // 16.8MB output from evicting embedding rows out of the 192MB L2.
    v4f* op = (v4f*)(out + (size_t)b * (T_TABLES * D_DIM)
                         + t * D_DIM + ((unsigned)lane << 2));
    __builtin_nontemporal_store(acc, op);
}

extern "C" void kernel_launch(void* const* d_in, const int* in_sizes, int n_in,
                              void* d_out, int out_size, void* d_ws, size_t ws_size,
                              hipStream_t stream) {
    (void)in_sizes; (void)n_in; (void)out_size; (void)d_ws; (void)ws_size;
    const float* tables  = (const float*)d_in[0];
    const int*   values  = (const int*)  d_in[1];
    const int*   offsets = (const int*)  d_in[2];
    const float* psw     = (const float*)d_in[3];
    float*       out     = (float*)d_out;

    // T*B = 32768 waves, 8 waves (256 threads) per block -> 4096 blocks.
    const int blocks = (T_TABLES * B_BAGS) / 8;
    grouped_embedding_bag_kernel<<<blocks, 256, 0, stream>>>(
        tables, values, offsets, psw, out);
}